// LocallyConnected2dV2_36369783062718
// MI455X (gfx1250) — compile-verified
//
#include <hip/hip_runtime.h>

typedef __attribute__((ext_vector_type(2))) float v2f;
typedef __attribute__((ext_vector_type(8))) float v8f;

#define IN_H 128
#define IN_W 128
#define KHH 8
#define KWW 8
#define OH 121
#define OW 121
#define BATCH 64
#define UW 24        // union footprint width: 16 + 8
#define KP 192       // expanded K: 8 rows * 24 cols
#define LSTRIDE 196  // padded LDS row stride (floats): 196%64==4 -> conflict-free B reads
#define NWAVES 4     // waves per block == batch tiles (64/16)

__global__ __launch_bounds__(128) void lc2d_wmma_f32_kernel(
    const float* __restrict__ x,      // [64,128,128]
    const float* __restrict__ w,      // [121,121,8,8]
    const float* __restrict__ bias,   // [121,121]
    float* __restrict__ out)          // [64,121,121]
{
    __shared__ float Wl[16 * LSTRIDE];          // expanded weights A' [16 x 192]
    __shared__ float Xl[NWAVES][16 * LSTRIDE];  // per-wave union patch, [batch 16 x 192]

    const int i    = blockIdx.x >> 3;        // output row, 0..120
    const int j0   = (blockIdx.x & 7) << 4;  // output col tile base
    const int tid  = threadIdx.x;
    const int wave = tid >> 5;               // batch tile 0..3
    const int lane = tid & 31;

    // ---- cooperative: zero + scatter-fill expanded weight tile (shared by all 4 waves)
    for (int idx = tid; idx < 16 * LSTRIDE; idx += 128) Wl[idx] = 0.0f;
    __syncthreads();
    for (int idx = tid; idx < 16 * (KHH * KWW); idx += 128) {
        const int jl = idx >> 6;            // local output col 0..15
        const int t  = idx & 63;            // kh*8+kw
        const int kh = t >> 3, kw = t & 7;
        const int j  = j0 + jl;
        if (j < OW) {
            // A'[jl, kh*24 + jl + kw] = w[i, j, kh, kw]
            Wl[jl * LSTRIDE + kh * UW + jl + kw] = w[(i * OW + j) * (KHH * KWW) + t];
        }
    }

    // ---- per-wave: load 8x24 union input window for 16 batches (row-major per batch)
    for (int idx = lane; idx < 16 * KP; idx += 32) {
        const int b   = idx / KP;
        const int k   = idx % KP;
        const int kh  = k / UW;
        const int c   = k % UW;
        const int col = j0 + c;
        float v = 0.0f;
        if (col < IN_W)
            v = x[((wave * 16 + b) * IN_H + (i + kh)) * IN_W + col];
        Xl[wave][b * LSTRIDE + k] = v;
    }
    __syncthreads();

    // ---- D[16 j x 16 batch] = A'[16 x 192] * X[192 x 16], 48 steps of f32 16x16x4 WMMA
    // A layout: lanes 0-15 -> M=0..15 with K=base+0/+1 in V0/V1; lanes 16-31 -> K=base+2/+3
    // B layout: lanes 0-15 -> N=0..15 with K=base+0/+1;          lanes 16-31 -> K=base+2/+3
    const int mn    = lane & 15;   // M row (for A) == N col/batch (for B)
    const int khalf = lane >> 4;   // selects K pair 0/1 vs 2/3
    const float* arow = &Wl[mn * LSTRIDE + khalf * 2];
    const float* brow = &Xl[wave][mn * LSTRIDE + khalf * 2];

    v8f acc = {};
    #pragma unroll
    for (int kk = 0; kk < KP / 4; ++kk) {
        v2f a = *(const v2f*)(arow + kk * 4);   // ds_load_b64 (8B aligned)
        v2f b = *(const v2f*)(brow + kk * 4);   // ds_load_b64 (8B aligned)
        acc = __builtin_amdgcn_wmma_f32_16x16x4_f32(
            /*neg_a=*/false, a, /*neg_b=*/false, b,
            /*c_mod=*/(short)0, acc, /*reuse_a=*/false, /*reuse_b=*/false);
    }

    // ---- store: VGPR v holds D[M = v + khalf*8, N = lane&15]
    const int bglob = wave * 16 + mn;
    #pragma unroll
    for (int v = 0; v < 8; ++v) {
        const int j = j0 + v + khalf * 8;
        if (j < OW)
            out[(bglob * OH + i) * OW + j] = acc[v] + bias[i * OW + j];
    }
}

extern "C" void kernel_launch(void* const* d_in, const int* in_sizes, int n_in,
                              void* d_out, int out_size, void* d_ws, size_t ws_size,
                              hipStream_t stream) {
    const float* x    = (const float*)d_in[0];
    const float* w    = (const float*)d_in[1];
    const float* bias = (const float*)d_in[2];
    float* out        = (float*)d_out;

    // 121 output rows x 8 column tiles of 16 (ceil(121/16)); 128 threads = 4 waves,
    // one wave per 16-batch tile (64 batches total).
    dim3 grid(OH * 8);
    dim3 block(128);
    lc2d_wmma_f32_kernel<<<grid, block, 0, stream>>>(x, w, bias, out);
}